// SiameseNodeFeaturesToEdgeFeatures_2052994367862
// MI455X (gfx1250) — compile-verified
//
#include <hip/hip_runtime.h>
#include <hip/hip_bf16.h>

// N=50000, D=128, E=800000, H=400000 (derived at runtime from in_sizes).
//
// Roofline:
//  - edge_attr = x[src]-x[dst]: 410MB of HBM writes -> ~18us floor @23.3TB/s.
//    b128 coalesced gathers (x is 25.6MB, L2-resident) + non-temporal b128
//    stores so the output stream doesn't evict x from L2.
//  - hyperedge cosines: 614MB of L2-resident gathers + trivial FLOPs.
//    CDNA5 async global->LDS staging (ASYNCcnt), wave32 shfl_xor reduction.
//  - FUSED into one persistent kernel: even waves do edge units, odd waves do
//    hyper units, so HBM-write-bound and L2-read-bound work overlap instead of
//    running as two serialized kernel phases. Total ~ max(T_edge, T_hyper).
// No matmul structure -> WMMA would waste >90% of every 16x16 tile; the
// gfx1250-specific path exercised here is the async-tensor/LDS data mover.

typedef float v4f __attribute__((ext_vector_type(4)));

#define D 128
#define WPB 8  // waves per 256-thread block (wave32)

static __device__ __forceinline__ float dot4(v4f a, v4f b) {
  return a.x * b.x + a.y * b.y + a.z * b.z + a.w * b.w;
}

__global__ __launch_bounds__(256) void fused_kernel(
    const float* __restrict__ x, const int* __restrict__ ei,
    const int* __restrict__ hi, float* __restrict__ out_e,
    float* __restrict__ out_h, int E, int H, int nwaves) {
  __shared__ __align__(16) float lds[WPB][2][3][D];  // 24 KB per block

  const int lane = threadIdx.x & 31;
  const int w = __builtin_amdgcn_readfirstlane((int)(threadIdx.x >> 5));
  const int gw = __builtin_amdgcn_readfirstlane((int)blockIdx.x) * WPB + w;

  const int EW = (E + 3) >> 2;  // edge units: 4 edges each
  const int HU = (H + 1) >> 1;  // hyper units: 2 hyperedges each
  const int M = (EW > HU) ? EW : HU;
  const int total = 2 * M;  // even unit -> edge, odd unit -> hyper

  // Stage one hyperedge's three 512B rows into LDS via async copies
  // (16B per lane per row). Tracked by ASYNCcnt; completes in order.
  auto stage = [&](int h, int buf) {
    const int ia = hi[h];  // h wave-uniform -> scalar s_load
    const int ib = hi[H + h];
    const int ic = hi[2 * H + h];
    const unsigned long long g0 =
        (unsigned long long)(x + (size_t)ia * D + lane * 4);
    const unsigned long long g1 =
        (unsigned long long)(x + (size_t)ib * D + lane * 4);
    const unsigned long long g2 =
        (unsigned long long)(x + (size_t)ic * D + lane * 4);
    const unsigned l0 = (unsigned)(unsigned long long)&lds[w][buf][0][lane * 4];
    const unsigned l1 = (unsigned)(unsigned long long)&lds[w][buf][1][lane * 4];
    const unsigned l2 = (unsigned)(unsigned long long)&lds[w][buf][2][lane * 4];
    asm volatile("global_load_async_to_lds_b128 %0, %1, off" ::"v"(l0), "v"(g0)
                 : "memory");
    asm volatile("global_load_async_to_lds_b128 %0, %1, off" ::"v"(l1), "v"(g1)
                 : "memory");
    asm volatile("global_load_async_to_lds_b128 %0, %1, off" ::"v"(l2), "v"(g2)
                 : "memory");
  };

  // Six dots + wave reduction + three cosines for one staged hyperedge.
  auto cosines = [&](int h, int buf) {
    const v4f u0 = *(const v4f*)&lds[w][buf][0][lane * 4];
    const v4f u1 = *(const v4f*)&lds[w][buf][1][lane * 4];
    const v4f u2 = *(const v4f*)&lds[w][buf][2][lane * 4];
    const v4f v01 = u0 - u1;
    const v4f v02 = u0 - u2;
    const v4f v12 = u1 - u2;

    float s11 = dot4(v01, v01);
    float s22 = dot4(v02, v02);
    float s33 = dot4(v12, v12);
    float s12 = dot4(v01, v02);
    float s13 = dot4(v01, v12);
    float s23 = dot4(v12, v02);
#pragma unroll
    for (int off = 16; off > 0; off >>= 1) {
      s11 += __shfl_xor(s11, off, 32);
      s22 += __shfl_xor(s22, off, 32);
      s33 += __shfl_xor(s33, off, 32);
      s12 += __shfl_xor(s12, off, 32);
      s13 += __shfl_xor(s13, off, 32);
      s23 += __shfl_xor(s23, off, 32);
    }
    const float n1 = sqrtf(s11), n2 = sqrtf(s22), n3 = sqrtf(s33);
    const float c1 = s12 / (n1 * n2);
    const float c2 = -s13 / (n1 * n3);
    const float c3 = s23 / (n3 * n2);
    if (lane < 3) {
      const float v = (lane == 0) ? c1 : (lane == 1) ? c2 : c3;
      out_h[(size_t)h * 3 + lane] = v;
    }
  };

  for (int u = gw; u < total; u += nwaves) {
    const int idx = u >> 1;
    if ((u & 1) == 0) {
      // ---------------- edge unit: 4 edges ----------------
      if (idx >= EW) continue;
      const int e0 = idx * 4;
#pragma unroll
      for (int j = 0; j < 4; ++j) {
        const int e = e0 + j;
        if (e >= E) break;
        const int s = __builtin_amdgcn_readfirstlane(ei[e]);
        const int d = __builtin_amdgcn_readfirstlane(ei[E + e]);
        const v4f a = *(const v4f*)(x + (size_t)s * D + lane * 4);
        const v4f b = *(const v4f*)(x + (size_t)d * D + lane * 4);
        const v4f r = a - b;
        __builtin_nontemporal_store(r,
                                    (v4f*)(out_e + (size_t)e * D + lane * 4));
      }
    } else {
      // ------------- hyper unit: 2 hyperedges, pipelined -------------
      if (idx >= HU) continue;
      const int h0 = idx * 2;
      const bool two = (h0 + 1) < H;
      stage(h0, 0);
      if (two) {
        stage(h0 + 1, 1);
        asm volatile("s_wait_asynccnt 3" ::: "memory");  // h0's rows done
      } else {
        asm volatile("s_wait_asynccnt 0" ::: "memory");
      }
      cosines(h0, 0);
      if (two) {
        asm volatile("s_wait_asynccnt 0" ::: "memory");  // h1's rows done
        cosines(h0 + 1, 1);
      }
    }
  }
}

extern "C" void kernel_launch(void* const* d_in, const int* in_sizes, int n_in,
                              void* d_out, int out_size, void* d_ws,
                              size_t ws_size, hipStream_t stream) {
  const float* x = (const float*)d_in[0];
  const int* ei = (const int*)d_in[1];
  const int* hi = (const int*)d_in[2];
  const int E = in_sizes[1] / 2;
  const int H = in_sizes[2] / 3;

  float* out_edge = (float*)d_out;
  float* out_hyper = out_edge + (size_t)E * D;

  // Persistent fused kernel; even/odd waves split edge vs hyper work so both
  // memory systems (HBM write stream, L2 gather stream) are driven at once.
  const int blocks = 8192;
  const int nwaves = blocks * WPB;
  fused_kernel<<<blocks, 256, 0, stream>>>(x, ei, hi, out_edge, out_hyper, E,
                                           H, nwaves);
}